// HCGAT_57294863729408
// MI455X (gfx1250) — compile-verified
//
#include <hip/hip_runtime.h>
#include <math.h>

typedef __attribute__((ext_vector_type(2))) float v2f;
typedef __attribute__((ext_vector_type(8))) float v8f;

#define NEG_SLOPE 0.2f

__device__ __forceinline__ float leaky(float e) { return e > 0.f ? e : NEG_SLOPE * e; }

// Map float -> uint such that uint ordering == float ordering (for atomicMax)
__device__ __forceinline__ unsigned ord_enc(float f) {
    unsigned u = __float_as_uint(f);
    return (u & 0x80000000u) ? ~u : (u | 0x80000000u);
}
__device__ __forceinline__ float ord_dec(unsigned u) {
    unsigned b = (u & 0x80000000u) ? (u & 0x7fffffffu) : ~u;
    return __uint_as_float(b);
}

// ---------------------------------------------------------------------------
// GEMM1: H[N,64] = X[N,12] @ W[12,64] using V_WMMA_F32_16X16X4_F32
// One wave (32 threads) per 16x16 output tile; 3 K-steps of 4.
// A 16x4 f32 layout (ISA table): lanes 0-15 rows, VGPR0/1 = K0/K1;
// lanes 16-31 = K2/K3. C/D: VGPR v -> row m0+v (+8 for upper half-wave).
// ---------------------------------------------------------------------------
__global__ __launch_bounds__(32) void hcgat_gemm1_wmma(
    const float* __restrict__ X, const float* __restrict__ W,
    float* __restrict__ H, int N) {
    int m0 = blockIdx.x * 16;
    int n0 = blockIdx.y * 16;
    int lane = threadIdx.x;
    int row = m0 + (lane & 15);
    int rowc = row < N ? row : N - 1;
    int col = n0 + (lane & 15);
    int khalf = (lane >> 4) * 2;  // 0 or 2
    v8f c = {};
#pragma unroll
    for (int k0 = 0; k0 < 12; k0 += 4) {
        v2f a, b;
        a[0] = X[(size_t)rowc * 12 + k0 + khalf + 0];
        a[1] = X[(size_t)rowc * 12 + k0 + khalf + 1];
        b[0] = W[(k0 + khalf + 0) * 64 + col];
        b[1] = W[(k0 + khalf + 1) * 64 + col];
        c = __builtin_amdgcn_wmma_f32_16x16x4_f32(false, a, false, b, (short)0, c,
                                                  false, false);
    }
    int rbase = m0 + (lane >> 4) * 8;
    if (m0 + 15 < N) {  // uniform fast path: no divergence, coalesced stores
#pragma unroll
        for (int v = 0; v < 8; ++v)
            H[(size_t)(rbase + v) * 64 + col] = c[v];
    } else {
#pragma unroll
        for (int v = 0; v < 8; ++v)
            if (rbase + v < N) H[(size_t)(rbase + v) * 64 + col] = c[v];
    }
}

// ---------------------------------------------------------------------------
// GEMM2: H[N,26] = X[N,64] @ W[64,26]  (cols padded to 32 -> two 16-tiles)
// ---------------------------------------------------------------------------
__global__ __launch_bounds__(32) void hcgat_gemm2_wmma(
    const float* __restrict__ X, const float* __restrict__ W,
    float* __restrict__ H, int N) {
    int m0 = blockIdx.x * 16;
    int n0 = blockIdx.y * 16;
    int lane = threadIdx.x;
    int row = m0 + (lane & 15);
    int rowc = row < N ? row : N - 1;
    int col = n0 + (lane & 15);
    int khalf = (lane >> 4) * 2;
    bool colok = (col < 26);
    v8f c = {};
#pragma unroll
    for (int k0 = 0; k0 < 64; k0 += 4) {
        v2f a, b;
        a[0] = X[(size_t)rowc * 64 + k0 + khalf + 0];
        a[1] = X[(size_t)rowc * 64 + k0 + khalf + 1];
        b[0] = colok ? W[(k0 + khalf + 0) * 26 + col] : 0.f;
        b[1] = colok ? W[(k0 + khalf + 1) * 26 + col] : 0.f;
        c = __builtin_amdgcn_wmma_f32_16x16x4_f32(false, a, false, b, (short)0, c,
                                                  false, false);
    }
    int rbase = m0 + (lane >> 4) * 8;
    if (m0 + 15 < N) {
        if (colok) {
#pragma unroll
            for (int v = 0; v < 8; ++v)
                H[(size_t)(rbase + v) * 26 + col] = c[v];
        }
    } else {
#pragma unroll
        for (int v = 0; v < 8; ++v)
            if (colok && rbase + v < N) H[(size_t)(rbase + v) * 26 + col] = c[v];
    }
}

// ---------------------------------------------------------------------------
// Per-node attention scores (layer 1, F=32/head), self-loop init of
// emax/denom, and zero the aggregation buffer.
// ---------------------------------------------------------------------------
__global__ void hcgat_scores1(const float* __restrict__ h1,
                              const float* __restrict__ a1s,
                              const float* __restrict__ a1d,
                              float* __restrict__ ssrc, float* __restrict__ sdst,
                              unsigned* __restrict__ emax, float* __restrict__ denom,
                              float* __restrict__ agg, int N) {
    int n = blockIdx.x * blockDim.x + threadIdx.x;
    if (n >= N) return;
    const float* hp = h1 + (size_t)n * 64;
    float* ap = agg + (size_t)n * 64;
    float ss0 = 0.f, ss1 = 0.f, sd0 = 0.f, sd1 = 0.f;
#pragma unroll
    for (int f = 0; f < 32; ++f) {
        float v0 = hp[f], v1 = hp[32 + f];
        ss0 += v0 * a1s[f];       sd0 += v0 * a1d[f];
        ss1 += v1 * a1s[32 + f];  sd1 += v1 * a1d[32 + f];
        ap[f] = 0.f; ap[32 + f] = 0.f;
    }
    ssrc[n * 2 + 0] = ss0; ssrc[n * 2 + 1] = ss1;
    sdst[n * 2 + 0] = sd0; sdst[n * 2 + 1] = sd1;
    emax[n * 2 + 0] = ord_enc(leaky(ss0 + sd0));   // self-loop edge seeds the max
    emax[n * 2 + 1] = ord_enc(leaky(ss1 + sd1));
    denom[n * 2 + 0] = 0.f; denom[n * 2 + 1] = 0.f;
}

__global__ void hcgat_scores2(const float* __restrict__ h2,
                              const float* __restrict__ a2s,
                              const float* __restrict__ a2d,
                              float* __restrict__ ssrc, float* __restrict__ sdst,
                              unsigned* __restrict__ emax, float* __restrict__ denom,
                              float* __restrict__ agg, int N) {
    int n = blockIdx.x * blockDim.x + threadIdx.x;
    if (n >= N) return;
    const float* hp = h2 + (size_t)n * 26;
    float* ap = agg + (size_t)n * 26;
    float ss0 = 0.f, ss1 = 0.f, sd0 = 0.f, sd1 = 0.f;
#pragma unroll
    for (int f = 0; f < 13; ++f) {
        float v0 = hp[f], v1 = hp[13 + f];
        ss0 += v0 * a2s[f];       sd0 += v0 * a2d[f];
        ss1 += v1 * a2s[13 + f];  sd1 += v1 * a2d[13 + f];
        ap[f] = 0.f; ap[13 + f] = 0.f;
    }
    ssrc[n * 2 + 0] = ss0; ssrc[n * 2 + 1] = ss1;
    sdst[n * 2 + 0] = sd0; sdst[n * 2 + 1] = sd1;
    emax[n * 2 + 0] = ord_enc(leaky(ss0 + sd0));
    emax[n * 2 + 1] = ord_enc(leaky(ss1 + sd1));
    denom[n * 2 + 0] = 0.f; denom[n * 2 + 1] = 0.f;
}

// ---------------------------------------------------------------------------
// Edge pass 1: segment max of leakyrelu(e) via ordered-uint atomicMax
// ---------------------------------------------------------------------------
__global__ void hcgat_edge_max(const int* __restrict__ src, const int* __restrict__ dst,
                               const float* __restrict__ ssrc,
                               const float* __restrict__ sdst,
                               unsigned* __restrict__ emax, int E) {
    int i = blockIdx.x * blockDim.x + threadIdx.x;
    if (i >= E) return;
    int s = src[i], d = dst[i];
#pragma unroll
    for (int h = 0; h < 2; ++h) {
        float e = leaky(ssrc[s * 2 + h] + sdst[d * 2 + h]);
        atomicMax(&emax[d * 2 + h], ord_enc(e));
    }
}

// ---------------------------------------------------------------------------
// Edge pass 2: p = exp(e - emax[dst]); denom[dst]+=p; agg[dst]+=p*h[src]
// (native f32 atomics via unsafeAtomicAdd -> global_atomic_add_f32)
// F=32: per-head block is 128B and 16B-aligned -> float4 (b128) loads.
// F=13: rows are 104B -> scalar loads.
// ---------------------------------------------------------------------------
template <int F>
__global__ void hcgat_edge_sum(const int* __restrict__ src, const int* __restrict__ dst,
                               const float* __restrict__ ssrc,
                               const float* __restrict__ sdst,
                               const unsigned* __restrict__ emax,
                               const float* __restrict__ h,
                               float* __restrict__ denom, float* __restrict__ agg,
                               int E) {
    int i = blockIdx.x * blockDim.x + threadIdx.x;
    if (i >= E) return;
    int s = src[i], d = dst[i];
    const float* hs = h + (size_t)s * (2 * F);
    float* ad = agg + (size_t)d * (2 * F);
    __builtin_prefetch(hs, 0, 0);
#pragma unroll
    for (int hh = 0; hh < 2; ++hh) {
        float e = leaky(ssrc[s * 2 + hh] + sdst[d * 2 + hh]);
        float p = __expf(e - ord_dec(emax[d * 2 + hh]));
        unsafeAtomicAdd(&denom[d * 2 + hh], p);
        if constexpr ((F & 3) == 0) {
            const float4* hv = reinterpret_cast<const float4*>(hs + hh * F);
            float* adh = ad + hh * F;
#pragma unroll
            for (int q = 0; q < F / 4; ++q) {
                float4 v = hv[q];
                unsafeAtomicAdd(&adh[4 * q + 0], p * v.x);
                unsafeAtomicAdd(&adh[4 * q + 1], p * v.y);
                unsafeAtomicAdd(&adh[4 * q + 2], p * v.z);
                unsafeAtomicAdd(&adh[4 * q + 3], p * v.w);
            }
        } else {
#pragma unroll
            for (int f = 0; f < F; ++f)
                unsafeAtomicAdd(&ad[hh * F + f], p * hs[hh * F + f]);
        }
    }
}

// ---------------------------------------------------------------------------
// Finalize layer 1: fold self-loop contribution, normalize, +bias, ReLU
// (in place in agg1 -> becomes input to GEMM2)
// ---------------------------------------------------------------------------
__global__ void hcgat_finalize1(const float* __restrict__ ssrc,
                                const float* __restrict__ sdst,
                                const unsigned* __restrict__ emax,
                                const float* __restrict__ denom,
                                const float* __restrict__ h1,
                                const float* __restrict__ b1,
                                float* __restrict__ agg, int N) {
    int n = blockIdx.x * blockDim.x + threadIdx.x;
    if (n >= N) return;
    float p[2], dn[2];
#pragma unroll
    for (int h = 0; h < 2; ++h) {
        float e = leaky(ssrc[n * 2 + h] + sdst[n * 2 + h]);
        p[h] = __expf(e - ord_dec(emax[n * 2 + h]));
        dn[h] = 1.f / (denom[n * 2 + h] + p[h] + 1e-16f);
    }
    const float* hp = h1 + (size_t)n * 64;
    float* ap = agg + (size_t)n * 64;
#pragma unroll
    for (int c = 0; c < 64; ++c) {
        int h = c >> 5;
        float v = (ap[c] + p[h] * hp[c]) * dn[h] + b1[c];
        ap[c] = v > 0.f ? v : 0.f;
    }
}

// ---------------------------------------------------------------------------
// Final: layer-2 normalize, mean over heads, +bias, sigmoid, hierarchy max
// ---------------------------------------------------------------------------
__global__ void hcgat_final(const float* __restrict__ ssrc,
                            const float* __restrict__ sdst,
                            const unsigned* __restrict__ emax,
                            const float* __restrict__ denom,
                            const float* __restrict__ h2,
                            const float* __restrict__ agg,
                            const float* __restrict__ b2,
                            const float* __restrict__ R,
                            float* __restrict__ out, int N) {
    int n = blockIdx.x * blockDim.x + threadIdx.x;
    if (n >= N) return;
    float p[2], dn[2];
#pragma unroll
    for (int h = 0; h < 2; ++h) {
        float e = leaky(ssrc[n * 2 + h] + sdst[n * 2 + h]);
        p[h] = __expf(e - ord_dec(emax[n * 2 + h]));
        dn[h] = 1.f / (denom[n * 2 + h] + p[h] + 1e-16f);
    }
    const float* hp = h2 + (size_t)n * 26;
    const float* ap = agg + (size_t)n * 26;
    float sg[13];
#pragma unroll
    for (int c = 0; c < 13; ++c) {
        float v0 = (ap[c] + p[0] * hp[c]) * dn[0];
        float v1 = (ap[13 + c] + p[1] * hp[13 + c]) * dn[1];
        float m = 0.5f * (v0 + v1) + b2[c];
        sg[c] = 1.f / (1.f + __expf(-m));
    }
#pragma unroll
    for (int i = 0; i < 13; ++i) {
        float mx = -1e30f;
#pragma unroll
        for (int j = 0; j < 13; ++j) {
            float t = R[i * 13 + j] * sg[j];
            mx = t > mx ? t : mx;
        }
        out[(size_t)n * 13 + i] = mx;
    }
}

extern "C" void kernel_launch(void* const* d_in, const int* in_sizes, int n_in,
                              void* d_out, int out_size, void* d_ws, size_t ws_size,
                              hipStream_t stream) {
    const float* x   = (const float*)d_in[0];
    const int*   ei  = (const int*)d_in[1];
    const float* R   = (const float*)d_in[2];
    const float* W1  = (const float*)d_in[3];
    const float* a1s = (const float*)d_in[4];
    const float* a1d = (const float*)d_in[5];
    const float* b1  = (const float*)d_in[6];
    const float* W2  = (const float*)d_in[7];
    const float* a2s = (const float*)d_in[8];
    const float* a2d = (const float*)d_in[9];
    const float* b2  = (const float*)d_in[10];

    int N = in_sizes[0] / 12;
    int E = in_sizes[1] / 2;
    const int* src = ei;
    const int* dst = ei + E;

    float* ws = (float*)d_ws;
    size_t off = 0;
    float*    h1     = ws + off; off += (size_t)N * 64;
    float*    s1src  = ws + off; off += (size_t)N * 2;
    float*    s1dst  = ws + off; off += (size_t)N * 2;
    unsigned* emax1  = (unsigned*)(ws + off); off += (size_t)N * 2;
    float*    denom1 = ws + off; off += (size_t)N * 2;
    float*    agg1   = ws + off; off += (size_t)N * 64;
    float*    h2     = ws + off; off += (size_t)N * 26;
    float*    s2src  = ws + off; off += (size_t)N * 2;
    float*    s2dst  = ws + off; off += (size_t)N * 2;
    unsigned* emax2  = (unsigned*)(ws + off); off += (size_t)N * 2;
    float*    denom2 = ws + off; off += (size_t)N * 2;
    float*    agg2   = ws + off; off += (size_t)N * 26;

    dim3 wave(32);
    int nb = (N + 255) / 256;
    int eb = (E + 255) / 256;

    // ---- layer 1 ----
    hcgat_gemm1_wmma<<<dim3((N + 15) / 16, 4), wave, 0, stream>>>(x, W1, h1, N);
    hcgat_scores1<<<nb, 256, 0, stream>>>(h1, a1s, a1d, s1src, s1dst, emax1, denom1,
                                          agg1, N);
    hcgat_edge_max<<<eb, 256, 0, stream>>>(src, dst, s1src, s1dst, emax1, E);
    hcgat_edge_sum<32><<<eb, 256, 0, stream>>>(src, dst, s1src, s1dst, emax1, h1,
                                               denom1, agg1, E);
    hcgat_finalize1<<<nb, 256, 0, stream>>>(s1src, s1dst, emax1, denom1, h1, b1, agg1,
                                            N);
    // ---- layer 2 ----
    hcgat_gemm2_wmma<<<dim3((N + 15) / 16, 2), wave, 0, stream>>>(agg1, W2, h2, N);
    hcgat_scores2<<<nb, 256, 0, stream>>>(h2, a2s, a2d, s2src, s2dst, emax2, denom2,
                                          agg2, N);
    hcgat_edge_max<<<eb, 256, 0, stream>>>(src, dst, s2src, s2dst, emax2, E);
    hcgat_edge_sum<13><<<eb, 256, 0, stream>>>(src, dst, s2src, s2dst, emax2, h2,
                                               denom2, agg2, E);
    hcgat_final<<<nb, 256, 0, stream>>>(s2src, s2dst, emax2, denom2, h2, agg2, b2, R,
                                        (float*)d_out, N);
}